// HOGGeneratorMel_65034394796196
// MI455X (gfx1250) — compile-verified
//
#include <hip/hip_runtime.h>
#include <math.h>

#define NBINS 9
#define POOL 8
#define GW 16
#define IMG 256
#define STRIP_ROWS 32          // pixel rows handled per workgroup
#define CELLS_R 4              // cell rows per strip
#define CELLS_C 32             // cell cols
#define THREADS 128            // 4 wave32s; one thread per cell
#define TILE_ROWS 34           // 32 rows + 1 halo row each side
#define LDS_STRIDE 264         // floats per tile row; pixel col c (-1..256) -> index 4+c
                               // 264*4 bytes = multiple of 16 -> interior (offset 16) stays 16B aligned

#if defined(__gfx1250__) && __has_builtin(__builtin_amdgcn_global_load_async_to_lds_b128)
#define USE_ASYNC_LDS 1
#else
#define USE_ASYNC_LDS 0
#endif

#if USE_ASYNC_LDS
typedef int v4i __attribute__((vector_size(16)));
typedef __attribute__((address_space(1))) v4i gv4i_t;   // prints as "int __vector __device__*"
typedef __attribute__((address_space(3))) v4i lv4i_t;   // prints as "int __vector __shared__*"
#endif

__device__ __forceinline__ int reflect_idx(int i) {
    // jnp.pad(..., mode='reflect') with pad=1: -1 -> 1, 256 -> 254
    if (i < 0) i = -i;
    if (i > IMG - 1) i = 2 * (IMG - 1) - i;
    return i;
}

__global__ __launch_bounds__(THREADS)
void hog_kernel(const float* __restrict__ x, float* __restrict__ out) {
    __shared__ __align__(16) float tile[TILE_ROWS * LDS_STRIDE];
    __shared__ float hist[THREADS * NBINS];
    __shared__ float gwin[GW];

    const int tid   = threadIdx.x;
    const int strip = blockIdx.x;          // 0..7
    const int b     = blockIdx.y;          // batch
    const int row0  = strip * STRIP_ROWS;  // first pixel row of this strip
    const float* img = x + (size_t)b * IMG * IMG;

    // ---- Gaussian window table (1-D); 2-D tiled kernel = w[r%16]*w[c%16]/S^2 ----
    float S = 0.0f;
    #pragma unroll
    for (int i = 0; i < GW; ++i) {
        float n = ((float)i - 7.5f) * 0.125f;   // (i - mean)/std, std = 8
        S += expf(-0.5f * n * n);
    }
    const float invS2 = 1.0f / (S * S);
    if (tid < GW) {
        float n = ((float)tid - 7.5f) * 0.125f;
        gwin[tid] = expf(-0.5f * n * n);
    }

    // ---- Stage 34 x 258 reflect-padded tile into LDS ----
    // Interior columns 0..255 as 16B chunks: 34 rows * 64 chunks = 2176 (17 per thread)
    for (int idx = tid; idx < TILE_ROWS * (IMG / 4); idx += THREADS) {
        const int r  = idx / (IMG / 4);
        const int c4 = idx % (IMG / 4);
        const int gr = reflect_idx(row0 - 1 + r);
        const float* gsrc = img + (size_t)gr * IMG + 4 * c4;
        float* ldst = &tile[r * LDS_STRIDE + 4 + 4 * c4];
#if USE_ASYNC_LDS
        v4i* gp_generic = (v4i*)const_cast<float*>(gsrc);  // pointee change, generic AS
        v4i* lp_generic = (v4i*)ldst;
        gv4i_t* gp = (gv4i_t*)gp_generic;                  // pure addrspace conversion -> AS1
        lv4i_t* lp = (lv4i_t*)lp_generic;                  // pure addrspace conversion -> AS3
        __builtin_amdgcn_global_load_async_to_lds_b128(gp, lp, 0, 0);
#else
        *(float4*)ldst = *(const float4*)gsrc;
#endif
    }
    // Halo columns: pixel col -1 -> col 1, pixel col 256 -> col 254
    if (tid < 2 * TILE_ROWS) {
        const int r    = tid >> 1;
        const int gr   = reflect_idx(row0 - 1 + r);
        if ((tid & 1) == 0) tile[r * LDS_STRIDE + 3]   = img[(size_t)gr * IMG + 1];
        else                tile[r * LDS_STRIDE + 260] = img[(size_t)gr * IMG + 254];
    }
#if USE_ASYNC_LDS
#if __has_builtin(__builtin_amdgcn_s_wait_asynccnt)
    __builtin_amdgcn_s_wait_asynccnt(0);
#else
    asm volatile("s_wait_asynccnt 0" ::: "memory");
#endif
#endif
    __syncthreads();

    // ---- Each thread owns one 8x8 cell: private 9-bin LDS histogram ----
    const int cr = tid >> 5;        // cell row within strip: 0..3
    const int cc = tid & 31;        // cell col: 0..31
    float* myh = &hist[tid * NBINS];
    #pragma unroll
    for (int k = 0; k < NBINS; ++k) myh[k] = 0.0f;

    const int prow_base = cr * POOL;   // local pixel row within strip
    const int pcol_base = cc * POOL;
    const float bin_scale = 9.0f * 0.318309886183790671538f;  // NBINS / pi

    for (int i = 0; i < POOL; ++i) {
        const int lr   = prow_base + i + 1;       // tile row (tile row 0 = pixel row row0-1)
        const int grow = row0 + prow_base + i;    // absolute pixel row (gaussian phase)
        const float wr = gwin[grow & (GW - 1)] * invS2;
        const float* t0 = &tile[(lr - 1) * LDS_STRIDE + 4];
        const float* t1 = &tile[(lr    ) * LDS_STRIDE + 4];
        const float* t2 = &tile[(lr + 1) * LDS_STRIDE + 4];
        #pragma unroll
        for (int j = 0; j < POOL; ++j) {
            const int c = pcol_base + j;
            const float a00 = t0[c - 1], a01 = t0[c], a02 = t0[c + 1];
            const float a10 = t1[c - 1],               a12 = t1[c + 1];
            const float a20 = t2[c - 1], a21 = t2[c], a22 = t2[c + 1];
            // cross-correlation with wt=[[1,0,-1],[2,0,-2],[1,0,-1]] and its transpose
            const float gx = (a00 - a02) + 2.0f * (a10 - a12) + (a20 - a22);
            const float gy = (a00 + 2.0f * a01 + a02) - (a20 + 2.0f * a21 + a22);
            const float mag = sqrtf(gx * gx + gy * gy);
            const float ph  = atan2f(gx, gy) * bin_scale;     // atan2(gx, gy)/pi*9
            int bin = (int)floorf(ph) % NBINS;
            if (bin < 0) bin += NBINS;
            const float wgt = wr * gwin[c & (GW - 1)];
            myh[bin] += mag * wgt;
        }
    }

    // ---- L2 normalize over bins, write [B, 32*32, 9] ----
    float nrm2 = 0.0f;
    #pragma unroll
    for (int k = 0; k < NBINS; ++k) { const float v = myh[k]; nrm2 += v * v; }
    const float inv = 1.0f / fmaxf(sqrtf(nrm2), 1e-12f);
    const int cellF = strip * CELLS_R + cr;
    float* o = out + ((size_t)b * (CELLS_C * CELLS_C) + (size_t)cellF * CELLS_C + cc) * NBINS;
    #pragma unroll
    for (int k = 0; k < NBINS; ++k) o[k] = myh[k] * inv;
}

extern "C" void kernel_launch(void* const* d_in, const int* in_sizes, int n_in,
                              void* d_out, int out_size, void* d_ws, size_t ws_size,
                              hipStream_t stream) {
    (void)n_in; (void)out_size; (void)d_ws; (void)ws_size;
    const float* x = (const float*)d_in[0];
    float* out = (float*)d_out;
    const int B = in_sizes[0] / (IMG * IMG);   // 64 for the reference shapes
    if (B <= 0) return;
    dim3 grid(IMG / STRIP_ROWS, B);            // (8 strips, B batches)
    hog_kernel<<<grid, THREADS, 0, stream>>>(x, out);
}